// KNNFeature_36438502540027
// MI455X (gfx1250) — compile-verified
//
#include <hip/hip_runtime.h>
#include <hip/hip_bf16.h>

typedef float v2f __attribute__((ext_vector_type(2)));
typedef float v8f __attribute__((ext_vector_type(8)));

#define NSLICE 32          // B*L
#define NPTS   4096
#define MQ     1024
#define KNN    16
#define CMID   64
#define COUT   128
#define R_TOTAL (NSLICE*MQ*KNN)   // 524288 rows of gathered neighbors
#define BN_EPS 1e-5f

// ---------------- workspace layout (floats) ----------------
#define OFF_G      0                       // [R_TOTAL*3] gathered coords
#define OFF_SUM1   (R_TOTAL*3)
#define OFF_SQ1    (OFF_SUM1 + CMID)
#define OFF_W1F    (OFF_SQ1  + CMID)       // folded W1 [64*3]
#define OFF_B1F    (OFF_W1F  + CMID*3)     // folded b1 [64]
#define OFF_SUM2   (OFF_B1F  + CMID)
#define OFF_SQ2    (OFF_SUM2 + COUT)
#define OFF_SCL2   (OFF_SQ2  + COUT)
#define OFF_SHF2   (OFF_SCL2 + COUT)

// ---------------- zero the stat accumulators ----------------
__global__ __launch_bounds__(256) void zero_stats_kernel(float* __restrict__ ws) {
  int t = threadIdx.x;
  if (t < CMID) { ws[OFF_SUM1 + t] = 0.f; ws[OFF_SQ1 + t] = 0.f; }
  if (t < COUT) { ws[OFF_SUM2 + t] = 0.f; ws[OFF_SQ2 + t] = 0.f; }
}

// ---------------- KNN + gather: one thread per query ----------------
// blockIdx: 128 blocks; 4 blocks per (b,l) slice.
__global__ __launch_bounds__(256) void knn_gather_kernel(const float* __restrict__ points,
                                                         const float* __restrict__ queries,
                                                         float* __restrict__ g) {
  __shared__ float lp[NPTS * 3];   // 48 KB: whole slice's points
  const int bl    = blockIdx.x >> 2;
  const int qbase = (blockIdx.x & 3) * 256;
  const float* p = points + (size_t)bl * NPTS * 3;
  for (int i = threadIdx.x; i < NPTS * 3; i += 256) lp[i] = p[i];
  __syncthreads();

  const int m = qbase + threadIdx.x;
  const float* q = queries + ((size_t)bl * MQ + m) * 3;
  const float qx = q[0], qy = q[1], qz = q[2];

  float dk[KNN];
  int   ik[KNN];
#pragma unroll
  for (int t = 0; t < KNN; ++t) { dk[t] = 3.4e38f; ik[t] = 0; }

  for (int j = 0; j < NPTS; ++j) {
    const float dx = lp[3*j + 0] - qx;
    const float dy = lp[3*j + 1] - qy;
    const float dz = lp[3*j + 2] - qz;
    const float d  = fmaf(dx, dx, fmaf(dy, dy, dz * dz));
    if (d < dk[KNN - 1]) {             // insert + bubble (stable on ties)
      dk[KNN - 1] = d; ik[KNN - 1] = j;
#pragma unroll
      for (int t = KNN - 1; t > 0; --t) {
        if (dk[t] < dk[t - 1]) {
          float td = dk[t]; dk[t] = dk[t - 1]; dk[t - 1] = td;
          int   ti = ik[t]; ik[t] = ik[t - 1]; ik[t - 1] = ti;
        }
      }
    }
  }

  float* go = g + ((size_t)bl * MQ + m) * KNN * 3;
#pragma unroll
  for (int t = 0; t < KNN; ++t) {
    const int idx = ik[t];
    go[t*3 + 0] = lp[3*idx + 0];
    go[t*3 + 1] = lp[3*idx + 1];
    go[t*3 + 2] = lp[3*idx + 2];
  }
}

// ---------------- BN1 statistics: sum/sumsq of h1 = g@W1^T + b1 ----------------
__global__ __launch_bounds__(256) void bn1_stats_kernel(const float* __restrict__ g,
                                                        const float* __restrict__ W1,
                                                        const float* __restrict__ b1,
                                                        float* __restrict__ ws) {
  const int c   = threadIdx.x & (CMID - 1);
  const int sub = threadIdx.x >> 6;            // 0..3 rows per block-iteration
  const float w0 = W1[c*3+0], w1 = W1[c*3+1], w2 = W1[c*3+2], b = b1[c];
  float s = 0.f, ss = 0.f;
  for (int r = blockIdx.x * 4 + sub; r < R_TOTAL; r += gridDim.x * 4) {
    const float x = g[r*3+0], y = g[r*3+1], z = g[r*3+2];
    const float h = fmaf(w0, x, fmaf(w1, y, fmaf(w2, z, b)));
    s += h; ss = fmaf(h, h, ss);
  }
  atomicAdd(&ws[OFF_SUM1 + c], s);
  atomicAdd(&ws[OFF_SQ1  + c], ss);
}

// ---------------- fold BN1 into W1/b1 ----------------
__global__ __launch_bounds__(64) void fold1_kernel(const float* __restrict__ W1,
                                                   const float* __restrict__ b1,
                                                   const float* __restrict__ gamma1,
                                                   const float* __restrict__ beta1,
                                                   float* __restrict__ ws) {
  const int c = threadIdx.x;
  const float inv_n = 1.0f / (float)R_TOTAL;
  const float mean  = ws[OFF_SUM1 + c] * inv_n;
  const float var   = fmaf(-mean, mean, ws[OFF_SQ1 + c] * inv_n);
  const float a     = gamma1[c] * rsqrtf(var + BN_EPS);
  ws[OFF_W1F + c*3 + 0] = a * W1[c*3 + 0];
  ws[OFF_W1F + c*3 + 1] = a * W1[c*3 + 1];
  ws[OFF_W1F + c*3 + 2] = a * W1[c*3 + 2];
  ws[OFF_B1F + c]       = fmaf(a, (b1[c] - mean), beta1[c]);
}

// ---------------- Linear1(folded,ReLU) -> Linear2 via WMMA f32 16x16x4 ----------------
// 256 threads = 8 waves; each wave: one 16-row strip x 128 output cols.
__global__ __launch_bounds__(256) void lin2_wmma_kernel(const float* __restrict__ g,
                                                        const float* __restrict__ W2,
                                                        const float* __restrict__ b2,
                                                        const float* __restrict__ ws,
                                                        float* __restrict__ h2out) {
  __shared__ float lw2[COUT * CMID];   // 32 KB
  __shared__ float lb2[COUT];
  __shared__ float lw1[CMID * 3];
  __shared__ float lb1[CMID];

  const int tid = threadIdx.x;
  for (int i = tid; i < COUT * CMID; i += 256) lw2[i] = W2[i];
  if (tid < COUT)     lb2[tid] = b2[tid];
  if (tid < CMID * 3) lw1[tid] = ws[OFF_W1F + tid];
  if (tid < CMID)     lb1[tid] = ws[OFF_B1F + tid];
  __syncthreads();

  const int wave  = tid >> 5;
  const int lane  = tid & 31;
  const int hi    = lane >> 4;     // 0: K even-pair, 1: K odd-pair (+2)
  const int lm    = lane & 15;
  const int strip = blockIdx.x * 8 + wave;
  const int row0  = strip * 16;

  // each lane holds coords of row (row0+lm); both half-waves mirror rows 0..15
  const float* gr = g + (size_t)(row0 + lm) * 3;
  const float gx = gr[0], gy = gr[1], gz = gr[2];

  // A fragments: af[kc] = relu(lin1) for channels c0, c0+1  (c0 = 4*kc + 2*hi)
  v2f af[16];
#pragma unroll
  for (int kc = 0; kc < 16; ++kc) {
    const int c0 = kc * 4 + hi * 2;
    const float h0 = fmaf(lw1[c0*3+0], gx, fmaf(lw1[c0*3+1], gy, fmaf(lw1[c0*3+2], gz, lb1[c0])));
    const float h1 = fmaf(lw1[(c0+1)*3+0], gx, fmaf(lw1[(c0+1)*3+1], gy, fmaf(lw1[(c0+1)*3+2], gz, lb1[c0+1])));
    af[kc].x = fmaxf(h0, 0.f);
    af[kc].y = fmaxf(h1, 0.f);
  }

  for (int nt = 0; nt < 8; ++nt) {
    const int col = nt * 16 + lm;
    const float bias = lb2[col];
    v8f acc;
#pragma unroll
    for (int v = 0; v < 8; ++v) acc[v] = bias;

#pragma unroll
    for (int kc = 0; kc < 16; ++kc) {
      const int k0 = kc * 4 + hi * 2;                       // even -> 8B aligned
      const v2f bf = *(const v2f*)(&lw2[col * CMID + k0]);  // B frag: W2[col][k0], W2[col][k0+1]
      acc = __builtin_amdgcn_wmma_f32_16x16x4_f32(
          /*neg_a=*/false, af[kc], /*neg_b=*/false, bf,
          /*c_mod=*/(short)0, acc, /*reuse_a=*/false, /*reuse_b=*/false);
    }

    float* o = h2out + (size_t)row0 * COUT + col;
#pragma unroll
    for (int v = 0; v < 8; ++v) {
      const int rloc = v + hi * 8;                          // C/D layout: M = v + 8*(lane>=16)
      o[(size_t)rloc * COUT] = acc[v];
    }
  }
}

// ---------------- BN2 statistics over h2 ----------------
__global__ __launch_bounds__(256) void bn2_stats_kernel(const float* __restrict__ h2,
                                                        float* __restrict__ ws) {
  const int c   = threadIdx.x & (COUT - 1);
  const int sub = threadIdx.x >> 7;            // 0..1
  float s = 0.f, ss = 0.f;
  for (int r = blockIdx.x * 2 + sub; r < R_TOTAL; r += gridDim.x * 2) {
    const float h = h2[(size_t)r * COUT + c];
    s += h; ss = fmaf(h, h, ss);
  }
  atomicAdd(&ws[OFF_SUM2 + c], s);
  atomicAdd(&ws[OFF_SQ2  + c], ss);
}

__global__ __launch_bounds__(128) void fold2_kernel(const float* __restrict__ gamma2,
                                                    const float* __restrict__ beta2,
                                                    float* __restrict__ ws) {
  const int c = threadIdx.x;
  const float inv_n = 1.0f / (float)R_TOTAL;
  const float mean  = ws[OFF_SUM2 + c] * inv_n;
  const float var   = fmaf(-mean, mean, ws[OFF_SQ2 + c] * inv_n);
  const float a     = gamma2[c] * rsqrtf(var + BN_EPS);
  ws[OFF_SCL2 + c] = a;
  ws[OFF_SHF2 + c] = fmaf(-mean, a, beta2[c]);
}

// ---------------- in-place BN2 + ReLU on d_out ----------------
__global__ __launch_bounds__(256) void bn2_apply_kernel(float* __restrict__ h2,
                                                        const float* __restrict__ ws) {
  const size_t i  = ((size_t)blockIdx.x * 256 + threadIdx.x) * 4;   // 4 consecutive channels
  const int    cb = (int)(i & (COUT - 1));
  float4 v = *(float4*)(&h2[i]);
  v.x = fmaxf(fmaf(v.x, ws[OFF_SCL2 + cb + 0], ws[OFF_SHF2 + cb + 0]), 0.f);
  v.y = fmaxf(fmaf(v.y, ws[OFF_SCL2 + cb + 1], ws[OFF_SHF2 + cb + 1]), 0.f);
  v.z = fmaxf(fmaf(v.z, ws[OFF_SCL2 + cb + 2], ws[OFF_SHF2 + cb + 2]), 0.f);
  v.w = fmaxf(fmaf(v.w, ws[OFF_SCL2 + cb + 3], ws[OFF_SHF2 + cb + 3]), 0.f);
  *(float4*)(&h2[i]) = v;
}

extern "C" void kernel_launch(void* const* d_in, const int* in_sizes, int n_in,
                              void* d_out, int out_size, void* d_ws, size_t ws_size,
                              hipStream_t stream) {
  const float* points  = (const float*)d_in[0];
  const float* queries = (const float*)d_in[1];
  const float* W1      = (const float*)d_in[2];
  const float* b1      = (const float*)d_in[3];
  const float* gamma1  = (const float*)d_in[4];
  const float* beta1   = (const float*)d_in[5];
  const float* W2      = (const float*)d_in[6];
  const float* b2      = (const float*)d_in[7];
  const float* gamma2  = (const float*)d_in[8];
  const float* beta2   = (const float*)d_in[9];

  float* ws  = (float*)d_ws;
  float* g   = ws + OFF_G;
  float* out = (float*)d_out;   // h2 staged here, then BN2+ReLU applied in place

  zero_stats_kernel<<<1, 256, 0, stream>>>(ws);
  knn_gather_kernel<<<NSLICE * MQ / 256, 256, 0, stream>>>(points, queries, g);
  bn1_stats_kernel<<<256, 256, 0, stream>>>(g, W1, b1, ws);
  fold1_kernel<<<1, 64, 0, stream>>>(W1, b1, gamma1, beta1, ws);
  lin2_wmma_kernel<<<R_TOTAL / (16 * 8), 256, 0, stream>>>(g, W2, b2, ws, out);
  bn2_stats_kernel<<<512, 256, 0, stream>>>(out, ws);
  fold2_kernel<<<1, 128, 0, stream>>>(gamma2, beta2, ws);
  bn2_apply_kernel<<<(R_TOTAL * COUT / 4) / 256, 256, 0, stream>>>(out, ws);
}